// PriorGuidedClustering_8366596293039
// MI455X (gfx1250) — compile-verified
//
#include <hip/hip_runtime.h>
#include <math.h>
#include <stdint.h>

// ---------------------------------------------------------------------------
// PriorGuidedClustering for MI455X (gfx1250, wave32, WMMA f32 16x16x4,
// async global->LDS staging with double buffering).
//
// Algebraic optimization: since C=8 << N=400, the k/v projections
// (2 x 53.7 GFLOP, 2 x 105MB intermediates) are factored through the
// prototype side:
//   scores[b,(h,c),n] = refined[b,n,:] . A2[(h,c),:]   (A2 = q @ k_w per head)
//   ctx  = (w @ refined) @ v_w per head
// leaving ~450MB HBM traffic (~20us @ 23.3 TB/s) and ~15 GFLOP of f32 GEMMs,
// all run through a generic wave32 WMMA f32_16x16x4 kernel (exact f32).
// Tile staging uses GLOBAL_LOAD_ASYNC_TO_LDS_B128 (ASYNCcnt) when available,
// double-buffered so global traffic overlaps the WMMA stream.
// ---------------------------------------------------------------------------

typedef float v2f __attribute__((ext_vector_type(2)));
typedef float v8f __attribute__((ext_vector_type(8)));
typedef int v4i_ __attribute__((vector_size(16)));

#define B_ 64
#define N_ 400
#define D_ 1024
#define C_ 8
#define H_ 8
#define HD_ 128
#define DN_ 409600
#define SCALE_ATTN 0.08838834764831845f  // 1/sqrt(128)
#define SLOPE 0.01f
#define EPS_LN 1e-5f

#if defined(__has_builtin)
#if __has_builtin(__builtin_amdgcn_global_load_async_to_lds_b128) && \
    __has_builtin(__builtin_amdgcn_s_wait_asynccnt)
#define HAVE_ASYNC_LDS 1
#endif
#endif

#ifdef HAVE_ASYNC_LDS
__device__ __forceinline__ void async_g2l_b128(const float* g, float* l) {
  // 16 bytes, global -> LDS, no VGPR round trip; tracked by ASYNCcnt.
  // Builtin is typed over the b128 payload: (v4i AS1*, v4i AS3*, imm, imm).
  __builtin_amdgcn_global_load_async_to_lds_b128(
      (__attribute__((address_space(1))) v4i_*)(uintptr_t)g,
      (__attribute__((address_space(3))) v4i_*)(uintptr_t)l, 0, 0);
}
#endif

// ---------------------------------------------------------------------------
// Generic f32 WMMA GEMM:  C[z][m][n] = sum_k A[z][m][k] * B(k,n) + biases
//   flags bit0: B is [K x N] row-major (NN); else B is [N x K] row-major (NT)
//   flags bit1: LeakyReLU epilogue
// Block: 256 threads (8 waves). Block tile 64(M) x 128(N), K-step 32.
// Wave (wm = wave%4, wn = wave/4) computes a 16 x 64 tile (4 accumulators).
// M must be a multiple of 64. N, K arbitrary (zero-padded in LDS).
// LDS pitch 36 floats: 144B rows (16B aligned for b128 async writes) and
// conflict-free WMMA fragment reads across all 32 lanes.
// ---------------------------------------------------------------------------
#define TM 64
#define TN 128
#define TK 32
#define LP 36

__device__ __forceinline__ void gemm_compute_tile(const float* As,
                                                  const float* Bs, v8f acc[4],
                                                  int wm, int wn, int ml,
                                                  int kh) {
#pragma unroll
  for (int ks = 0; ks < TK; ks += 4) {
    // A fragment: lane holds A(m=ml, k=ks+v+2*kh), v in {0,1}
    const int ar = (16 * wm + ml) * LP + ks + 2 * kh;
    v2f a; a.x = As[ar]; a.y = As[ar + 1];
#pragma unroll
    for (int j = 0; j < 4; ++j) {
      const int br = (64 * wn + 16 * j + ml) * LP + ks + 2 * kh;
      v2f b; b.x = Bs[br]; b.y = Bs[br + 1];
      acc[j] = __builtin_amdgcn_wmma_f32_16x16x4_f32(
          false, a, false, b, (short)0, acc[j], false, false);
    }
  }
}

#ifdef HAVE_ASYNC_LDS
__device__ __forceinline__ void gemm_issue_tile_async(
    const float* Ab, const float* Bb, float* Asb, float* Bsb, int m0, int n0,
    int K, long kO, int tid) {
  // A: 64 rows x 8 b128 chunks = 512 chunks / 256 threads
#pragma unroll
  for (int i = 0; i < 2; ++i) {
    int ch = tid + i * 256;
    int r = ch >> 3, kc = (ch & 7) << 2;
    async_g2l_b128(Ab + (long)(m0 + r) * K + kO + kc, Asb + r * LP + kc);
  }
  // B (NT): 128 rows x 8 b128 chunks = 1024 chunks / 256 threads
#pragma unroll
  for (int i = 0; i < 4; ++i) {
    int ch = tid + i * 256;
    int r = ch >> 3, kc = (ch & 7) << 2;
    async_g2l_b128(Bb + (long)(n0 + r) * K + kO + kc, Bsb + r * LP + kc);
  }
}
#endif

__global__ __launch_bounds__(256) void k_gemm_wmma(
    const float* __restrict__ A, const float* __restrict__ Bm,
    const float* __restrict__ biasN, const float* __restrict__ biasM,
    float* __restrict__ Cm, int M, int N, int K, int ldc,
    long sAb, long sBb, long sCb, int flags) {
  __shared__ float As[2][TM * LP];
  __shared__ float Bs[2][TN * LP];
  const int tid = threadIdx.x;
  const int n0 = blockIdx.x * TN;
  const int m0 = blockIdx.y * TM;
  const float* Ab = A + (long)blockIdx.z * sAb;
  const float* Bb = Bm + (long)blockIdx.z * sBb;
  float* Cb = Cm + (long)blockIdx.z * sCb;
  const int wave = tid >> 5, lane = tid & 31;
  const int wm = wave & 3, wn = wave >> 2;
  const int ml = lane & 15, kh = lane >> 4;
  const bool isNN = (flags & 1) != 0;
  const bool doRelu = (flags & 2) != 0;
  const bool alignedA = ((K & (TK - 1)) == 0);
  const bool alignedB = (!isNN) && alignedA && ((N & (TN - 1)) == 0);
  const int nT = (K + TK - 1) / TK;

  v8f acc[4];
  acc[0] = (v8f){}; acc[1] = (v8f){}; acc[2] = (v8f){}; acc[3] = (v8f){};

#ifdef HAVE_ASYNC_LDS
  if (alignedA && alignedB) {
    // Fully asynchronous, double-buffered pipeline: tile t+1 streams into the
    // other LDS buffer while tile t feeds the WMMA pipe.
    gemm_issue_tile_async(Ab, Bb, As[0], Bs[0], m0, n0, K, 0, tid);
    for (int t = 0; t < nT; ++t) {
      if (t == 0) {
        __builtin_amdgcn_s_wait_asynccnt(0);
        __syncthreads();
      }
      const int cur = t & 1;
      if (t + 1 < nT)
        gemm_issue_tile_async(Ab, Bb, As[cur ^ 1], Bs[cur ^ 1], m0, n0, K,
                              (long)(t + 1) * TK, tid);
      gemm_compute_tile(As[cur], Bs[cur], acc, wm, wn, ml, kh);
      __builtin_amdgcn_s_wait_asynccnt(0);
      __syncthreads();
    }
  } else
#endif
  {
    for (int t = 0; t < nT; ++t) {
      const int k0 = t * TK;
      // stage A tile (TM x TK)
#ifdef HAVE_ASYNC_LDS
      if (alignedA) {
#pragma unroll
        for (int i = 0; i < 2; ++i) {
          int ch = tid + i * 256;
          int r = ch >> 3, kc = (ch & 7) << 2;
          async_g2l_b128(Ab + (long)(m0 + r) * K + k0 + kc,
                         &As[0][r * LP + kc]);
        }
      } else
#endif
      {
        for (int i = 0; i < (TM * TK) / 256; ++i) {
          int e = tid + i * 256;
          int r = e >> 5, kk = e & 31;
          int kg = k0 + kk;
          As[0][r * LP + kk] = (kg < K) ? Ab[(long)(m0 + r) * K + kg] : 0.f;
        }
      }
      // stage B tile (TN x TK) as Bs[n][k], guarded (N and/or K tails)
      if (!isNN) {
        for (int i = 0; i < (TN * TK) / 256; ++i) {
          int e = tid + i * 256;
          int r = e >> 5, kk = e & 31;
          int ng = n0 + r, kg = k0 + kk;
          Bs[0][r * LP + kk] =
              (ng < N && kg < K) ? Bb[(long)ng * K + kg] : 0.f;
        }
      } else {
        for (int i = 0; i < (TN * TK) / 256; ++i) {
          int e = tid + i * 256;
          int kk = e >> 7, r = e & 127;  // coalesced along n
          int ng = n0 + r, kg = k0 + kk;
          Bs[0][r * LP + kk] =
              (ng < N && kg < K) ? Bb[(long)kg * N + ng] : 0.f;
        }
      }
      if (k0 + TK < K)
        __builtin_prefetch(Ab + (long)(m0 + (tid & 63)) * K + k0 + TK, 0, 1);
#ifdef HAVE_ASYNC_LDS
      if (alignedA) __builtin_amdgcn_s_wait_asynccnt(0);
#endif
      __syncthreads();
      gemm_compute_tile(As[0], Bs[0], acc, wm, wn, ml, kh);
      __syncthreads();
    }
  }

  // epilogue: VGPR r -> (m = r + 8*kh, n = ml) within 16x16 tile
#pragma unroll
  for (int j = 0; j < 4; ++j) {
    int n = n0 + 64 * wn + 16 * j + ml;
    if (n < N) {
      float bn = biasN ? biasN[n] : 0.f;
#pragma unroll
      for (int r = 0; r < 8; ++r) {
        int m = m0 + 16 * wm + r + 8 * kh;
        float v = acc[j][r] + bn;
        if (biasM) v += biasM[m];
        if (doRelu) v = (v >= 0.f) ? v : SLOPE * v;
        Cb[(long)m * ldc + n] = v;
      }
    }
  }
}

// --------------------------- dice prior normalize ---------------------------
__global__ void k_dice(const float* __restrict__ dp, float* __restrict__ dn) {
  int n = blockIdx.x * blockDim.x + threadIdx.x;
  if (n >= N_) return;
  float v[C_], s = 0.f;
#pragma unroll
  for (int c = 0; c < C_; ++c) { v[c] = dp[n * C_ + c]; s += v[c] * v[c]; }
  float inv = 1.f / fmaxf(sqrtf(s), 1e-12f);
#pragma unroll
  for (int c = 0; c < C_; ++c) dn[n * C_ + c] = v[c] * inv;
}

// ----------------- encoder layer 1 split-K partials (no atomics) ------------
// grid 400 blocks; each handles 8 k-tiles of 128; part[blk][64][32]
// LDS pitch 132 floats: 16B-aligned rows for b128 async staging.
#define EP 132
__global__ __launch_bounds__(256) void k_enc1(
    const float* __restrict__ flat, const float* __restrict__ w1,
    float* __restrict__ part) {
  __shared__ float w1t[32 * EP];
  __shared__ float ft[64 * EP];
  int tid = threadIdx.x;
  int nn = tid & 31, mg = tid >> 5;
  float acc[8];
#pragma unroll
  for (int j = 0; j < 8; ++j) acc[j] = 0.f;
  for (int it = 0; it < 8; ++it) {
    long kb = (long)blockIdx.x * 1024 + it * 128;
#ifdef HAVE_ASYNC_LDS
    // 32x128 + 64x128 floats as b128 chunks: 4 + 8 per thread
#pragma unroll
    for (int i = 0; i < 4; ++i) {
      int ch = tid + i * 256;
      int r = ch >> 5, kc = (ch & 31) << 2;
      async_g2l_b128(w1 + (long)r * DN_ + kb + kc, &w1t[r * EP + kc]);
    }
#pragma unroll
    for (int i = 0; i < 8; ++i) {
      int ch = tid + i * 256;
      int r = ch >> 5, kc = (ch & 31) << 2;
      async_g2l_b128(flat + (long)r * DN_ + kb + kc, &ft[r * EP + kc]);
    }
    __builtin_amdgcn_s_wait_asynccnt(0);
#else
    for (int i = 0; i < 16; ++i) {
      int e = tid + i * 256; int r = e >> 7, kk = e & 127;
      w1t[r * EP + kk] = w1[(long)r * DN_ + kb + kk];
    }
    for (int i = 0; i < 32; ++i) {
      int e = tid + i * 256; int r = e >> 7, kk = e & 127;
      ft[r * EP + kk] = flat[(long)r * DN_ + kb + kk];
    }
#endif
    __syncthreads();
    for (int kk = 0; kk < 128; ++kk) {
      float wv = w1t[nn * EP + kk];
#pragma unroll
      for (int j = 0; j < 8; ++j)
        acc[j] += wv * ft[(mg * 8 + j) * EP + kk];
    }
    __syncthreads();
  }
#pragma unroll
  for (int j = 0; j < 8; ++j)
    part[(long)blockIdx.x * 2048 + (mg * 8 + j) * 32 + nn] = acc[j];
}

__global__ void k_enc1red(const float* __restrict__ part, float* __restrict__ h1) {
  int idx = blockIdx.x * blockDim.x + threadIdx.x;
  if (idx >= 2048) return;
  float s = 0.f;
  for (int i = 0; i < 400; ++i) s += part[(long)i * 2048 + idx];
  h1[idx] = s;
}

// -------- encoder layers 1-activation + 2: h2 = lrelu(lrelu(h1+b1)@w2.T+b2) -
__global__ __launch_bounds__(256) void k_enc23(
    const float* __restrict__ h1, const float* __restrict__ b1,
    const float* __restrict__ w2, const float* __restrict__ b2,
    float* __restrict__ h2) {
  __shared__ float h1s[2048];
  int tid = threadIdx.x;
  for (int i = 0; i < 8; ++i) {
    int e = tid + i * 256;
    float v = h1[e] + b1[e & 31];
    h1s[e] = (v >= 0.f) ? v : SLOPE * v;
  }
  __syncthreads();
  for (int i = 0; i < 8; ++i) {
    int e = tid + i * 256;
    int m = e >> 5, n = e & 31;
    float a = b2[n];
#pragma unroll 8
    for (int j = 0; j < 32; ++j) a += h1s[m * 32 + j] * w2[n * 32 + j];
    h2[e] = (a >= 0.f) ? a : SLOPE * a;
  }
}

// ------------------- encoder layer 3: refined = h2 @ w3.T + b3 --------------
__global__ __launch_bounds__(256) void k_enc3(
    const float* __restrict__ h2, const float* __restrict__ w3,
    const float* __restrict__ b3, float* __restrict__ refined) {
  __shared__ float h2s[2048];
  int tid = threadIdx.x;
  for (int i = 0; i < 8; ++i) h2s[tid + i * 256] = h2[tid + i * 256];
  __syncthreads();
  long k = (long)blockIdx.x * 256 + tid;
  float r[32];
  const float4* wr = (const float4*)(w3 + k * 32);
#pragma unroll
  for (int i = 0; i < 8; ++i) {
    float4 f = wr[i];
    r[i * 4] = f.x; r[i * 4 + 1] = f.y; r[i * 4 + 2] = f.z; r[i * 4 + 3] = f.w;
  }
  float bias = b3[k];
  for (int m = 0; m < 64; ++m) {
    float a = bias;
#pragma unroll 8
    for (int j = 0; j < 32; ++j) a += h2s[m * 32 + j] * r[j];
    refined[(long)m * DN_ + k] = a;
  }
}

// ----------------------- q projection (tiny, C=8 rows) ----------------------
__global__ void k_qproj(const float* __restrict__ proto,
                        const float* __restrict__ qw,
                        const float* __restrict__ qb, float* __restrict__ q) {
  int idx = blockIdx.x * blockDim.x + threadIdx.x;
  if (idx >= C_ * D_) return;
  int c = idx >> 10, dout = idx & 1023;
  const float4* pr = (const float4*)(proto + c * D_);
  const float4* wr = (const float4*)(qw + (long)dout * D_);
  float a = qb[dout];
  for (int j = 0; j < 256; ++j) {
    float4 p = pr[j], w = wr[j];
    a += p.x * w.x + p.y * w.y + p.z * w.z + p.w * w.w;
  }
  q[idx] = a;
}

// ------------- A2[(h,c),d] = sum_dl q[c,h,dl] * k_w[h*128+dl, d] ------------
__global__ void k_a2(const float* __restrict__ q, const float* __restrict__ kw,
                     float* __restrict__ A2) {
  int idx = blockIdx.x * blockDim.x + threadIdx.x;  // 65536
  int m = idx >> 10, d = idx & 1023;
  int h = m >> 3, c = m & 7;
  const float* qrow = q + c * D_ + h * HD_;
  const float* kcol = kw + (long)(h * HD_) * D_ + d;
  float a = 0.f;
  for (int dl = 0; dl < HD_; ++dl) a += qrow[dl] * kcol[(long)dl * D_];
  A2[idx] = a;
}

// -------- sbias[(h,c)] = q[c,h,:] . k_b[h slice]  (k_b term of scores) ------
__global__ void k_sbias(const float* __restrict__ q, const float* __restrict__ kb,
                        float* __restrict__ sb) {
  int m = threadIdx.x;
  if (m >= 64) return;
  int h = m >> 3, c = m & 7;
  float a = 0.f;
  for (int dl = 0; dl < HD_; ++dl)
    a += q[c * D_ + h * HD_ + dl] * kb[h * HD_ + dl];
  sb[m] = a;
}

// ------------------ exact 1.5-entmax over n=400 (in-place) ------------------
// One 512-thread block per (b, h*8+c) row. Applies scale * dice bias, then
// the sort-based Peters&Martins algorithm. Also reduces wsum = sum_n p.
__global__ __launch_bounds__(512) void k_entmax(
    float* __restrict__ sc, const float* __restrict__ dn,
    float* __restrict__ wsum) {
  __shared__ float zo[512], sa[512], ta[512], cs[512], c2[512];
  int t = threadIdx.x;
  int b = blockIdx.x >> 6, m = blockIdx.x & 63;
  int c = m & 7;
  long base = (long)b * 25600 + (long)m * 400;
  float z = -3.0e38f;
  if (t < N_) z = sc[base + t] * SCALE_ATTN * dn[t * C_ + c] * 0.5f;
  zo[t] = z;
  ta[t] = z;
  __syncthreads();
  for (int s = 256; s > 0; s >>= 1) {
    if (t < s) ta[t] = fmaxf(ta[t], ta[t + s]);
    __syncthreads();
  }
  float zmax = ta[0];
  __syncthreads();
  sa[t] = z - zmax;
  __syncthreads();
  // bitonic sort, descending, 512 elems (tail padded with -inf-ish)
  for (int k = 2; k <= 512; k <<= 1) {
    for (int j = k >> 1; j > 0; j >>= 1) {
      int ixj = t ^ j;
      if (ixj > t) {
        float a = sa[t], bb = sa[ixj];
        bool up = ((t & k) == 0);
        bool sw = up ? (a < bb) : (a > bb);
        if (sw) { sa[t] = bb; sa[ixj] = a; }
      }
      __syncthreads();
    }
  }
  // inclusive scans of zs and zs^2 (Hillis-Steele)
  float sv = (t < N_) ? sa[t] : 0.f;
  cs[t] = sv; c2[t] = sv * sv;
  __syncthreads();
  for (int off = 1; off < 512; off <<= 1) {
    float a1 = 0.f, a2 = 0.f;
    if (t >= off) { a1 = cs[t - off]; a2 = c2[t - off]; }
    __syncthreads();
    cs[t] += a1; c2[t] += a2;
    __syncthreads();
  }
  float taut = -3.0e38f;
  if (t < N_) {
    float k1 = (float)(t + 1);
    float mean = cs[t] / k1, msq = c2[t] / k1;
    float ss = k1 * (msq - mean * mean);
    float delta = (1.f - ss) / k1;
    taut = mean - sqrtf(fmaxf(delta, 0.f));
  }
  ta[t] = taut;
  __syncthreads();
  cs[t] = (t < N_ && taut <= sa[t]) ? 1.f : 0.f;
  __syncthreads();
  for (int s = 256; s > 0; s >>= 1) {
    if (t < s) cs[t] += cs[t + s];
    __syncthreads();
  }
  int support = (int)cs[0];
  float taustar = ta[support - 1];
  __syncthreads();
  float p = 0.f;
  if (t < N_) {
    float v = fmaxf(zo[t] - zmax - taustar, 0.f);
    p = v * v;
    sc[base + t] = p;  // in-place: scores buffer becomes attn weights w
  }
  c2[t] = p;
  __syncthreads();
  for (int s = 256; s > 0; s >>= 1) {
    if (t < s) c2[t] += c2[t + s];
    __syncthreads();
  }
  if (t == 0) wsum[blockIdx.x] = c2[0];
}

// ---------------- soft_assignment[b,n,c] = mean_h w[b,(h,c),n] --------------
__global__ void k_soft(const float* __restrict__ w, float* __restrict__ out) {
  int idx = blockIdx.x * blockDim.x + threadIdx.x;
  if (idx >= B_ * N_ * C_) return;
  int b = idx / 3200, r = idx % 3200;
  int n = r >> 3, c = r & 7;
  float s = 0.f;
#pragma unroll
  for (int h = 0; h < H_; ++h) s += w[(long)b * 25600 + (h * 8 + c) * 400 + n];
  out[idx] = 0.125f * s;
}

// --- ctx + prototypes: xpre[b,c,d] = proto + T[b,(h(d),c),:] . v_w[d,:] + ...
__global__ __launch_bounds__(256) void k_ctx(
    const float* __restrict__ T, const float* __restrict__ vw,
    const float* __restrict__ vb, const float* __restrict__ wsum,
    const float* __restrict__ proto, float* __restrict__ xpre) {
  __shared__ float Tl[8 * 1024];
  __shared__ float wsl[8];
  int tid = threadIdx.x;
  int b = blockIdx.x >> 3, c = blockIdx.x & 7;
  for (int i = 0; i < 32; ++i) {
    int e = tid + i * 256;
    int h = e >> 10, d2 = e & 1023;
    Tl[e] = T[(long)b * 65536 + (long)(h * 8 + c) * 1024 + d2];
  }
  if (tid < 8) wsl[tid] = wsum[b * 64 + tid * 8 + c];
  __syncthreads();
  for (int i = 0; i < 4; ++i) {
    int d = i * 256 + tid;
    int h = d >> 7;
    float a = wsl[h] * vb[d] + proto[c * D_ + d];
    const float4* vr = (const float4*)(vw + (long)d * D_);
    const float* tr = Tl + h * 1024;
    for (int j = 0; j < 256; ++j) {
      float4 f = vr[j];
      a += f.x * tr[j * 4] + f.y * tr[j * 4 + 1] + f.z * tr[j * 4 + 2] +
           f.w * tr[j * 4 + 3];
    }
    xpre[(long)b * 8192 + c * 1024 + d] = a;
  }
}

// --------------------------- LayerNorm (optional residual) ------------------
__global__ __launch_bounds__(256) void k_ln(
    const float* __restrict__ a, const float* __restrict__ r,
    const float* __restrict__ g, const float* __restrict__ bt,
    float* __restrict__ o) {
  __shared__ float s1[256], s2[256];
  int tid = threadIdx.x;
  long base = (long)blockIdx.x * D_;
  float v[4], sum = 0.f, sq = 0.f;
#pragma unroll
  for (int i = 0; i < 4; ++i) {
    int d = tid + i * 256;
    float x = a[base + d];
    if (r) x += r[base + d];
    v[i] = x; sum += x; sq += x * x;
  }
  s1[tid] = sum; s2[tid] = sq;
  __syncthreads();
  for (int s = 128; s > 0; s >>= 1) {
    if (tid < s) { s1[tid] += s1[tid + s]; s2[tid] += s2[tid + s]; }
    __syncthreads();
  }
  float mu = s1[0] / (float)D_;
  float var = s2[0] / (float)D_ - mu * mu;
  float rs = rsqrtf(var + EPS_LN);
#pragma unroll
  for (int i = 0; i < 4; ++i) {
    int d = tid + i * 256;
    o[base + d] = (v[i] - mu) * rs * g[d] + bt[d];
  }
}

// -------------------- self-attention among C=8 communities ------------------
// one 128-thread block per (b, h)
__global__ __launch_bounds__(128) void k_attn(
    const float* __restrict__ qkv, float* __restrict__ aw,
    float* __restrict__ att) {
  __shared__ float ql[1024], kl[1024], vl[1024], sl[64];
  int tid = threadIdx.x;
  int b = blockIdx.x >> 3, h = blockIdx.x & 7;
  for (int i = 0; i < 8; ++i) {
    int e = tid + i * 128;
    int c = e >> 7, dl = e & 127;
    long base = (long)(b * 8 + c) * 3072 + h * HD_ + dl;
    ql[e] = qkv[base];
    kl[e] = qkv[base + 1024];
    vl[e] = qkv[base + 2048];
  }
  __syncthreads();
  if (tid < 64) {
    int c = tid >> 3, k2 = tid & 7;
    float s = 0.f;
    for (int dl = 0; dl < HD_; ++dl) s += ql[c * HD_ + dl] * kl[k2 * HD_ + dl];
    sl[tid] = s * SCALE_ATTN;
  }
  __syncthreads();
  if (tid < 8) {
    float mx = -3.0e38f;
#pragma unroll
    for (int k2 = 0; k2 < 8; ++k2) mx = fmaxf(mx, sl[tid * 8 + k2]);
    float sum = 0.f;
#pragma unroll
    for (int k2 = 0; k2 < 8; ++k2) {
      float e = __expf(sl[tid * 8 + k2] - mx);
      sl[tid * 8 + k2] = e; sum += e;
    }
    float inv = 1.f / sum;
#pragma unroll
    for (int k2 = 0; k2 < 8; ++k2) sl[tid * 8 + k2] *= inv;
  }
  __syncthreads();
  if (tid < 64) aw[(long)blockIdx.x * 64 + tid] = sl[tid];
  for (int i = 0; i < 8; ++i) {
    int e = tid + i * 128;
    int c = e >> 7, dl = e & 127;
    float s = 0.f;
#pragma unroll
    for (int k2 = 0; k2 < 8; ++k2) s += sl[c * 8 + k2] * vl[k2 * HD_ + dl];
    att[(long)(b * 8 + c) * D_ + h * HD_ + dl] = s;
  }
}

// ------------------- interaction[b,c,k] = mean_h aw[b,h,c,k] ----------------
__global__ void k_inter(const float* __restrict__ aw, float* __restrict__ out) {
  int idx = blockIdx.x * blockDim.x + threadIdx.x;
  if (idx >= B_ * C_ * C_) return;
  int b = idx >> 6, r = idx & 63;
  float s = 0.f;
#pragma unroll
  for (int h = 0; h < H_; ++h) s += aw[(long)(b * 8 + h) * 64 + r];
  out[idx] = 0.125f * s;
}

// ---------------------------------------------------------------------------
extern "C" void kernel_launch(void* const* d_in, const int* in_sizes, int n_in,
                              void* d_out, int out_size, void* d_ws,
                              size_t ws_size, hipStream_t stream) {
  const float* node  = (const float*)d_in[0];
  const float* dice  = (const float*)d_in[1];
  const float* proto = (const float*)d_in[2];
  const float* e_w1  = (const float*)d_in[3];
  const float* e_b1  = (const float*)d_in[4];
  const float* e_w2  = (const float*)d_in[5];
  const float* e_b2  = (const float*)d_in[6];
  const float* e_w3  = (const float*)d_in[7];
  const float* e_b3  = (const float*)d_in[8];
  const float* q_w   = (const float*)d_in[9];
  const float* q_b   = (const float*)d_in[10];
  const float* k_w   = (const float*)d_in[11];
  const float* k_b   = (const float*)d_in[12];
  const float* v_w   = (const float*)d_in[13];
  const float* v_b   = (const float*)d_in[14];
  const float* mi_w  = (const float*)d_in[15];
  const float* mi_b  = (const float*)d_in[16];
  const float* mo_w  = (const float*)d_in[17];
  const float* mo_b  = (const float*)d_in[18];
  const float* f_w1  = (const float*)d_in[19];
  const float* f_b1  = (const float*)d_in[20];
  const float* f_w2  = (const float*)d_in[21];
  const float* f_b2  = (const float*)d_in[22];
  const float* ln1g  = (const float*)d_in[23];
  const float* ln1b  = (const float*)d_in[24];
  const float* ln2g  = (const float*)d_in[25];
  const float* ln2b  = (const float*)d_in[26];
  const float* ln3g  = (const float*)d_in[27];
  const float* ln3b  = (const float*)d_in[28];
  float* out = (float*)d_out;

  // workspace layout (floats)
  float* ws = (float*)d_ws;
  size_t o = 0;
  float* REF  = ws + o; o += (size_t)26214400;  // refined [64,400,1024]
  float* PART = ws + o; o += (size_t)400 * 2048;
  float* H1   = ws + o; o += 2048;
  float* H2   = ws + o; o += 2048;
  float* QP   = ws + o; o += 8192;
  float* A2   = ws + o; o += 65536;
  float* SB   = ws + o; o += 64;
  float* DN   = ws + o; o += 3200;
  float* SC   = ws + o; o += (size_t)1638400;   // scores -> w (in place)
  float* WS   = ws + o; o += 4096;
  float* Tb   = ws + o; o += (size_t)4194304;   // T [64,64,1024]
  float* XPRE = ws + o; o += 524288;
  float* X1   = ws + o; o += 524288;
  float* QKV  = ws + o; o += (size_t)1572864;
  float* AW   = ws + o; o += 32768;
  float* ATT  = ws + o; o += 524288;
  float* SA   = ws + o; o += 524288;
  float* X2   = ws + o; o += 524288;
  float* FH   = ws + o; o += (size_t)2097152;
  float* FO   = ws + o; o += 524288;
  (void)ws_size; (void)in_sizes; (void)n_in; (void)out_size;

  // 1. dice prior L2-normalize
  k_dice<<<2, 256, 0, stream>>>(dice, DN);
  // 2-4. encoder
  k_enc1<<<400, 256, 0, stream>>>(node, e_w1, PART);
  k_enc1red<<<8, 256, 0, stream>>>(PART, H1);
  k_enc23<<<1, 256, 0, stream>>>(H1, e_b1, e_w2, e_b2, H2);
  k_enc3<<<1600, 256, 0, stream>>>(H2, e_w3, e_b3, REF);
  // 5. prototype-side precomputes
  k_qproj<<<32, 256, 0, stream>>>(proto, q_w, q_b, QP);
  k_a2<<<256, 256, 0, stream>>>(QP, k_w, A2);
  k_sbias<<<1, 64, 0, stream>>>(QP, k_b, SB);
  // 6. scores[b,(h,c),n] = A2 @ refined^T  (WMMA, NT, per-m bias; async-A)
  k_gemm_wmma<<<dim3(4, 1, 64), 256, 0, stream>>>(
      A2, REF, nullptr, SB, SC, 64, 400, 1024, 400,
      0L, (long)409600, (long)25600, 0);
  // 7. entmax-1.5 with dice bias (in place) + wsum
  k_entmax<<<4096, 512, 0, stream>>>(SC, DN, WS);
  // 8. soft assignment output
  k_soft<<<800, 256, 0, stream>>>(SC, out + 524288);
  // 9. T[b,(h,c),d'] = w @ refined  (WMMA, NN, guarded path)
  k_gemm_wmma<<<dim3(8, 1, 64), 256, 0, stream>>>(
      SC, REF, nullptr, nullptr, Tb, 64, 1024, 400, 1024,
      (long)25600, (long)409600, (long)65536, 1);
  // 10. ctx + prototypes, then LN1
  k_ctx<<<512, 256, 0, stream>>>(Tb, v_w, v_b, WS, proto, XPRE);
  k_ln<<<512, 256, 0, stream>>>(XPRE, nullptr, ln1g, ln1b, X1);
  // 11. qkv projection (WMMA, fully-async pipeline)
  k_gemm_wmma<<<dim3(24, 8, 1), 256, 0, stream>>>(
      X1, mi_w, mi_b, nullptr, QKV, 512, 3072, 1024, 3072, 0L, 0L, 0L, 0);
  // 12. tiny self-attention over C=8; interaction output
  k_attn<<<512, 128, 0, stream>>>(QKV, AW, ATT);
  k_inter<<<16, 256, 0, stream>>>(AW, out + 524288 + 204800);
  // 13. output projection (WMMA) + LN2
  k_gemm_wmma<<<dim3(8, 8, 1), 256, 0, stream>>>(
      ATT, mo_w, mo_b, nullptr, SA, 512, 1024, 1024, 1024, 0L, 0L, 0L, 0);
  k_ln<<<512, 256, 0, stream>>>(X1, SA, ln2g, ln2b, X2);
  // 14. FFN (WMMA x2, LeakyReLU fused in first) + LN3 -> out
  k_gemm_wmma<<<dim3(32, 8, 1), 256, 0, stream>>>(
      X2, f_w1, f_b1, nullptr, FH, 512, 4096, 1024, 4096, 0L, 0L, 0L, 2);
  k_gemm_wmma<<<dim3(8, 8, 1), 256, 0, stream>>>(
      FH, f_w2, f_b2, nullptr, FO, 512, 1024, 4096, 1024, 0L, 0L, 0L, 0);
  k_ln<<<512, 256, 0, stream>>>(X2, FO, ln3g, ln3b, out);
}